// LPrompt_91259465105703
// MI455X (gfx1250) — compile-verified
//
#include <hip/hip_runtime.h>

typedef float v2f __attribute__((ext_vector_type(2)));
typedef float v8f __attribute__((ext_vector_type(8)));

#define EMBED  768
#define HEADS  12
#define HEAD_D 64
#define TOPK   3
#define SEQ_N  197
#define PLEN   5

// ---------------------------------------------------------------------------
// block reduction over 192 threads; returns total to all threads
// ---------------------------------------------------------------------------
__device__ __forceinline__ float breduce192(volatile float* red, int t, float v) {
  red[t] = v; __syncthreads();
  if (t < 96) red[t] += red[t + 96]; __syncthreads();
  if (t < 48) red[t] += red[t + 48]; __syncthreads();
  if (t < 24) red[t] += red[t + 24]; __syncthreads();
  if (t < 12) red[t] += red[t + 12]; __syncthreads();
  if (t < 6)  red[t] += red[t + 6];  __syncthreads();
  if (t == 0) red[0] = red[0] + red[1] + red[2] + red[3] + red[4] + red[5];
  __syncthreads();
  float r = red[0];
  __syncthreads();
  return r;
}

// ---------------------------------------------------------------------------
// K0: key_norm[l,:] = l2_normalize(prompt_key[l,:]) for l < lmax
// grid = lmax blocks, 192 threads (one float4 column each)
// ---------------------------------------------------------------------------
__global__ __launch_bounds__(192) void keynorm_kernel(
    const float* __restrict__ pkey, const int* __restrict__ lmax_p,
    float* __restrict__ keyn) {
  const int l = blockIdx.x;
  if (l >= *lmax_p) return;
  const int t = threadIdx.x;
  __shared__ float red[192];
  const float* row = pkey + (size_t)l * EMBED + t * 4;
  float a0 = row[0], a1 = row[1], a2 = row[2], a3 = row[3];
  float tot = breduce192(red, t, a0*a0 + a1*a1 + a2*a2 + a3*a3);
  float rinv = rsqrtf(fmaxf(tot, 1e-12f));
  float* o = keyn + (size_t)l * EMBED + t * 4;
  o[0] = a0 * rinv; o[1] = a1 * rinv; o[2] = a2 * rinv; o[3] = a3 * rinv;
}

// ---------------------------------------------------------------------------
// K1: desc_embed[l,e] = sum_t new_desc_embed[l,t] * W_text[e,t]   (tiny GEMM)
// ---------------------------------------------------------------------------
__global__ __launch_bounds__(256) void desc_kernel(
    const float* __restrict__ nde, const float* __restrict__ Wtext,
    float* __restrict__ desc, int n) {
  const int idx = blockIdx.x * 256 + threadIdx.x;
  if (idx >= n) return;
  const int l = idx / EMBED, e = idx % EMBED;
  const float* a = nde + (size_t)l * EMBED;
  const float* w = Wtext + (size_t)e * EMBED;
  float acc = 0.f;
  for (int t = 0; t < EMBED; ++t) acc += a[t] * w[t];
  desc[idx] = acc;
}

// ---------------------------------------------------------------------------
// K2: per batch row b — stream mean over N=197, l2-norm, similarity[lmax],
// top-3, raw = sum_k topv[k]*desc_embed[topi[k]].  Dominant HBM kernel.
// grid = B blocks, 192 threads (one float4 column of E=768 each)
// ---------------------------------------------------------------------------
__global__ __launch_bounds__(192) void mean_sim_raw_kernel(
    const float* __restrict__ x_embed, const float* __restrict__ key_norm,
    const float* __restrict__ desc_embed, const int* __restrict__ lmax_p,
    float* __restrict__ sim_out, float* __restrict__ desc_out,
    float* __restrict__ sim_ws, float* __restrict__ raw_ws) {
  const int b = blockIdx.x;
  const int t = threadIdx.x;
  const int lmax = *lmax_p;
  __shared__ float red[192];
  __shared__ float simv[32];
  __shared__ float tv[TOPK];
  __shared__ int   ti[TOPK];

  const float* xb = x_embed + (size_t)b * SEQ_N * EMBED;
  float a0 = 0.f, a1 = 0.f, a2 = 0.f, a3 = 0.f;
  for (int n = 0; n < SEQ_N; ++n) {
    const float4 v = *(const float4*)(xb + (size_t)n * EMBED + t * 4);
    a0 += v.x; a1 += v.y; a2 += v.z; a3 += v.w;
  }
  const float inv_n = 1.0f / (float)SEQ_N;
  a0 *= inv_n; a1 *= inv_n; a2 *= inv_n; a3 *= inv_n;

  float tot = breduce192(red, t, a0*a0 + a1*a1 + a2*a2 + a3*a3);
  float rinv = rsqrtf(fmaxf(tot, 1e-12f));
  const float x0 = a0*rinv, x1 = a1*rinv, x2 = a2*rinv, x3 = a3*rinv;

  for (int l = 0; l < lmax; ++l) {
    const float* kn = key_norm + (size_t)l * EMBED + t * 4;
    float p = x0*kn[0] + x1*kn[1] + x2*kn[2] + x3*kn[3];
    float s = breduce192(red, t, p);
    if (t == 0) { simv[l] = s; sim_out[(size_t)b * lmax + l] = s; }
  }
  __syncthreads();

  if (t == 0) {  // top-3 (descending, ties -> lower index; matches lax.top_k)
    unsigned used = 0;
    for (int k = 0; k < TOPK; ++k) {
      float best = -INFINITY; int bi = 0;
      for (int l = 0; l < lmax; ++l)
        if (!((used >> l) & 1u) && simv[l] > best) { best = simv[l]; bi = l; }
      used |= 1u << bi; tv[k] = best; ti[k] = bi;
    }
  }
  __syncthreads();

  float r0 = 0.f, r1 = 0.f, r2 = 0.f, r3 = 0.f;
  for (int k = 0; k < TOPK; ++k) {
    const float* de = desc_embed + (size_t)ti[k] * EMBED + t * 4;
    const float w = tv[k];
    r0 += w * de[0]; r1 += w * de[1]; r2 += w * de[2]; r3 += w * de[3];
  }
  float4 rv; rv.x = r0; rv.y = r1; rv.z = r2; rv.w = r3;
  *(float4*)(desc_out + (size_t)b * EMBED + t * 4) = rv;
  *(float4*)(raw_ws  + (size_t)b * EMBED + t * 4) = rv;
  if (t < TOPK) sim_ws[(size_t)b * TOPK + t] = (t < lmax) ? simv[t] : 0.f;
}

// ---------------------------------------------------------------------------
// K3: proj[B,768] = raw[B,768] @ W_prompt^T  via V_WMMA_F32_16X16X4_F32
// one wave per 16x16 output tile; K-loop 768/4 = 192 wmma ops
// A layout: lane m = lane&15, k-halves by lane>>4 (2 f32/lane)
// B layout: lane n = lane&15, same k-halving; B[k][n] = W_prompt[tn+n][k]
// ---------------------------------------------------------------------------
__global__ __launch_bounds__(32) void proj_wmma_kernel(
    const float* __restrict__ rawm, const float* __restrict__ Wp,
    float* __restrict__ proj) {
  const int tn = blockIdx.x * 16;
  const int tm = blockIdx.y * 16;
  const int lane = threadIdx.x;
  const int half = lane >> 4;
  const int mr = lane & 15;
  v8f c = {};
  const float* arow = rawm + (size_t)(tm + mr) * EMBED;
  const float* brow = Wp   + (size_t)(tn + mr) * EMBED;
  for (int k = 0; k < EMBED; k += 4) {
    const int k0 = k + half * 2;
    v2f a, b;
    a.x = arow[k0]; a.y = arow[k0 + 1];
    b.x = brow[k0]; b.y = brow[k0 + 1];
    c = __builtin_amdgcn_wmma_f32_16x16x4_f32(false, a, false, b, (short)0, c,
                                              false, false);
  }
  for (int j = 0; j < 8; ++j)
    proj[(size_t)(tm + j + 8 * half) * EMBED + tn + mr] = c[j];
}

// ---------------------------------------------------------------------------
// K4: per (x,h) block: fold Cp = Wproj_p @ Wv_p into LDS, then WMMA
// Y_p = xh_h @ Cp^T (tiles), kv = sum_p sim[:,p] * (Y_p + bproj_p),
// replicate 5x into batched_prompt[b, x*5+l, h*64+d].
// (softmax over singleton == 1, so attention output == v; q,k are dead code)
// ---------------------------------------------------------------------------
__global__ __launch_bounds__(256) void head_kernel(
    const float* __restrict__ proj, const float* __restrict__ Wqkv,
    const float* __restrict__ Wproj, const float* __restrict__ bproj,
    const float* __restrict__ sim3, float* __restrict__ bp_out, int B) {
  const int xh = blockIdx.x;             // 0..23
  const int x = xh / HEADS, h = xh % HEADS;
  const int tid = threadIdx.x;
  __shared__ float Cp[TOPK][HEAD_D * HEAD_D];   // 48 KB

  const size_t base_q = (size_t)xh * TOPK * 192 * HEAD_D; // +p*12288 +(128+e)*64+f
  const size_t base_p = (size_t)xh * TOPK * HEAD_D * HEAD_D;
  const size_t base_b = (size_t)xh * TOPK * HEAD_D;

  // Cp[p][d*64+f] = sum_e Wproj[x,h,p,d,e] * Wqkv[x,h,p,128+e,f]
  for (int idx = tid; idx < TOPK * HEAD_D * HEAD_D; idx += 256) {
    const int p = idx >> 12;
    const int d = (idx >> 6) & 63;
    const int f = idx & 63;
    const float* wp = Wproj + base_p + (size_t)p * 4096 + d * 64;
    const float* wv = Wqkv + base_q + (size_t)p * 12288 + 128 * 64 + f;
    float acc = 0.f;
    for (int e = 0; e < HEAD_D; ++e) acc += wp[e] * wv[e * 64];
    Cp[p][d * 64 + f] = acc;
  }
  __syncthreads();

  const int wave = tid >> 5;
  const int lane = tid & 31;
  const int half = lane >> 4;
  const int mr = lane & 15;
  const int ntiles = (B / 16) * (HEAD_D / 16);   // 64 for B=256

  for (int t = wave; t < ntiles; t += 8) {
    const int tm = (t >> 2) * 16;        // batch tile
    const int tn = (t & 3) * 16;         // d tile
    float kvj[8] = {0.f, 0.f, 0.f, 0.f, 0.f, 0.f, 0.f, 0.f};
    const float* arow = proj + (size_t)(tm + mr) * EMBED + h * HEAD_D;
    for (int p = 0; p < TOPK; ++p) {
      v8f c = {};
      const float* brow = &Cp[p][(tn + mr) * 64];
      for (int k = 0; k < HEAD_D; k += 4) {
        const int k0 = k + half * 2;
        v2f a, b;
        a.x = arow[k0]; a.y = arow[k0 + 1];
        b.x = brow[k0]; b.y = brow[k0 + 1];
        c = __builtin_amdgcn_wmma_f32_16x16x4_f32(false, a, false, b, (short)0,
                                                  c, false, false);
      }
      const float bias = bproj[base_b + p * HEAD_D + tn + mr];
      for (int j = 0; j < 8; ++j) {
        const int brow_i = tm + j + 8 * half;
        kvj[j] += sim3[(size_t)brow_i * TOPK + p] * (c[j] + bias);
      }
    }
    const int d = tn + mr;
    for (int j = 0; j < 8; ++j) {
      const int brow_i = tm + j + 8 * half;
      float* o = bp_out + ((size_t)brow_i * (2 * PLEN) + x * PLEN) * EMBED +
                 h * HEAD_D + d;
      const float val = kvj[j];
      o[0] = val; o[EMBED] = val; o[2 * EMBED] = val; o[3 * EMBED] = val;
      o[4 * EMBED] = val;
    }
  }
}

// ---------------------------------------------------------------------------
extern "C" void kernel_launch(void* const* d_in, const int* in_sizes, int n_in,
                              void* d_out, int out_size, void* d_ws,
                              size_t ws_size, hipStream_t stream) {
  const float* x_embed   = (const float*)d_in[0];
  const float* prompt_key= (const float*)d_in[1];
  const float* W_text    = (const float*)d_in[2];
  const float* nde       = (const float*)d_in[3];
  const float* W_prompt  = (const float*)d_in[4];
  const float* Wqkv      = (const float*)d_in[5];
  const float* Wproj     = (const float*)d_in[6];
  const float* bproj     = (const float*)d_in[7];
  const int*   lmax_p    = (const int*)d_in[8];

  const int B  = in_sizes[0] / (SEQ_N * EMBED);              // 256
  const int LH = out_size / B - (EMBED + 2 * PLEN * EMBED);  // lmax (10)
  const int ND = in_sizes[3] / EMBED;                        // 10 desc rows

  float* out_sim  = (float*)d_out;                        // [B, lmax]
  float* out_desc = out_sim + (size_t)B * LH;             // [B, 768]
  float* out_bp   = out_desc + (size_t)B * EMBED;         // [B, 10, 768]

  float* key_ws  = (float*)d_ws;                          // [LH, 768]
  float* desc_ws = key_ws + (size_t)LH * EMBED;           // [ND, 768]
  float* sim_ws  = desc_ws + (size_t)ND * EMBED;          // [B, 3]
  float* raw_ws  = sim_ws + (size_t)B * TOPK;             // [B, 768]
  float* proj_ws = raw_ws + (size_t)B * EMBED;            // [B, 768]

  keynorm_kernel<<<LH, 192, 0, stream>>>(prompt_key, lmax_p, key_ws);
  desc_kernel<<<(ND * EMBED + 255) / 256, 256, 0, stream>>>(nde, W_text,
                                                            desc_ws, ND * EMBED);
  mean_sim_raw_kernel<<<B, 192, 0, stream>>>(x_embed, key_ws, desc_ws, lmax_p,
                                             out_sim, out_desc, sim_ws, raw_ws);
  dim3 g3(EMBED / 16, B / 16);
  proj_wmma_kernel<<<g3, 32, 0, stream>>>(raw_ws, W_prompt, proj_ws);
  head_kernel<<<2 * HEADS, 256, 0, stream>>>(proj_ws, Wqkv, Wproj, bproj,
                                             sim_ws, out_bp, B);
}